// MultiQueryAttention_35931696398603
// MI455X (gfx1250) — compile-verified
//
#include <hip/hip_runtime.h>

// ---------------------------------------------------------------------------
// MQA forward for MI455X (gfx1250): fp32 end-to-end using V_WMMA_F32_16X16X4_F32
// B=2 T=2048 D=2048 N=16 H=128, DQKV = D + 2H = 2304
// Transposed-in-LDS B/V operands (single ds_load_b64 per fragment),
// async global->LDS staging (ASYNCcnt) with the toolchain's v4i* signature,
// global prefetch hints for the next K tile.
// ---------------------------------------------------------------------------

#define B_    2
#define T_    2048
#define D_    2048
#define N_    16
#define H_    128
#define DQKV_ 2304

typedef __attribute__((ext_vector_type(2))) float v2f;
typedef __attribute__((ext_vector_type(8))) float v8f;

#if __has_builtin(__builtin_amdgcn_global_load_async_to_lds_b128) && \
    __has_builtin(__builtin_amdgcn_s_wait_asynccnt)
#define USE_ASYNC_LDS 1
// clang's declared param types: v4i in AS(1) for the global src,
// v4i in AS(3) for the LDS dst (per round-2 diagnostics).
typedef int v4i_ __attribute__((vector_size(16)));
typedef __attribute__((address_space(1))) v4i_ gv4i;
typedef __attribute__((address_space(3))) v4i_ lv4i;
#define ASYNC_B128(gsrc, ldst) \
    __builtin_amdgcn_global_load_async_to_lds_b128((gv4i*)(gsrc), (lv4i*)(ldst), 0, 0)
#else
#define USE_ASYNC_LDS 0
#endif

__device__ __forceinline__ v8f wmma_f32(v2f a, v2f b, v8f c) {
    // D = A(16x4) * B(4x16) + C(16x16), full fp32
    return __builtin_amdgcn_wmma_f32_16x16x4_f32(
        /*neg_a=*/false, a, /*neg_b=*/false, b,
        /*c_mod=*/(short)0, c, /*reuse_a=*/false, /*reuse_b=*/false);
}

// ---------------------------------------------------------------------------
// Generic fp32 WMMA GEMM: C[M,Ncol] = A[M,K] @ Bm[K,Ncol]
// 128x128 block tile, 256 threads (8 waves), each wave -> 32x64 (2x4 WMMA tiles)
// A staged row-major (async B128 when available); B staged TRANSPOSED so every
// B fragment is one aligned ds_load_b64.
// ---------------------------------------------------------------------------
__global__ __launch_bounds__(256) void gemm128_f32(const float* __restrict__ A,
                                                   const float* __restrict__ Bm,
                                                   float* __restrict__ C,
                                                   int M, int Ncol, int K) {
    __shared__ float As[128][36];    // 128x32, stride 36 (16B-aligned stores)
    __shared__ float Bst[128][34];   // B tile transposed: [col][k], stride even

    const int tid  = threadIdx.x;
    const int lane = tid & 31;
    const int wid  = tid >> 5;
    const int wr   = wid & 3;        // wave row group (32 rows each)
    const int wc   = wid >> 2;       // wave col group (64 cols each)
    const int lh   = lane >> 4;      // lane half
    const int ll   = lane & 15;

    const int rowBase = blockIdx.y * 128;
    const int colBase = blockIdx.x * 128;

    v8f acc[2][4] = {};

    for (int kt = 0; kt < K; kt += 32) {
        // stage A tile (128x32): 1024 x float4 over 256 threads
        #pragma unroll
        for (int i = 0; i < 4; ++i) {
            const int fi = tid + 256 * i;
            const int r  = fi >> 3;
            const int c4 = (fi & 7) << 2;
            const float* src = &A[(size_t)(rowBase + r) * K + kt + c4];
#if USE_ASYNC_LDS
            ASYNC_B128(src, &As[r][c4]);
#else
            *(float4*)&As[r][c4] = *(const float4*)src;
#endif
        }
        // stage B tile (32x128) transposed into Bst[col][k]
        #pragma unroll
        for (int i = 0; i < 4; ++i) {
            const int fi = tid + 256 * i;
            const int r  = fi >> 5;          // k within tile
            const int c4 = (fi & 31) << 2;   // col within tile
            const float4 v = *(const float4*)&Bm[(size_t)(kt + r) * Ncol + colBase + c4];
            Bst[c4 + 0][r] = v.x;
            Bst[c4 + 1][r] = v.y;
            Bst[c4 + 2][r] = v.z;
            Bst[c4 + 3][r] = v.w;
        }
        // prefetch next K tile into cache while we compute this one
        if (kt + 32 < K) {
            if (tid < 128) {
                __builtin_prefetch(&A[(size_t)(rowBase + tid) * K + kt + 32], 0, 1);
            } else {
                const int r = (tid - 128) >> 2;
                const int c = ((tid - 128) & 3) << 5;
                __builtin_prefetch(&Bm[(size_t)(kt + 32 + r) * Ncol + colBase + c], 0, 1);
            }
        }
#if USE_ASYNC_LDS
        __builtin_amdgcn_s_wait_asynccnt(0);
#endif
        __syncthreads();

        #pragma unroll
        for (int ks = 0; ks < 8; ++ks) {
            const int kk = ks * 4 + 2 * lh;   // this lane-half's K pair
            v2f a[2], b[4];
            #pragma unroll
            for (int t = 0; t < 2; ++t)
                a[t] = *(const v2f*)&As[wr * 32 + t * 16 + ll][kk];
            #pragma unroll
            for (int c = 0; c < 4; ++c)
                b[c] = *(const v2f*)&Bst[wc * 64 + c * 16 + ll][kk];
            #pragma unroll
            for (int t = 0; t < 2; ++t)
                #pragma unroll
                for (int c = 0; c < 4; ++c)
                    acc[t][c] = wmma_f32(a[t], b[c], acc[t][c]);
        }
        __syncthreads();
    }

    // epilogue: C/D layout -> row = tile + v + 8*laneHalf, col = tile + lane%16
    #pragma unroll
    for (int t = 0; t < 2; ++t)
        #pragma unroll
        for (int c = 0; c < 4; ++c)
            #pragma unroll
            for (int v = 0; v < 8; ++v) {
                const int row = rowBase + wr * 32 + t * 16 + v + 8 * lh;
                const int col = colBase + wc * 64 + c * 16 + ll;
                C[(size_t)row * Ncol + col] = acc[t][c][v];
            }
}

// ---------------------------------------------------------------------------
// RoPE applied in place to q heads (cols [0,D)) and the shared k (cols [D,D+H)).
// sin/cos rows are [ang, ang] so sin[t][j] == sin[t][j+64]; one pair per thread.
// ---------------------------------------------------------------------------
__global__ __launch_bounds__(256) void rope_kernel(float* __restrict__ qkv,
                                                   const float* __restrict__ sinT,
                                                   const float* __restrict__ cosT) {
    const int idx = blockIdx.x * 256 + threadIdx.x;   // exact grid, no tail
    const int j   = idx & 63;
    const int c   = (idx >> 6) % (N_ + 1);            // head 0..15, 16 == k
    const int bt  = idx / (64 * (N_ + 1));
    const int t   = bt % T_;

    float* p = qkv + (size_t)bt * DQKV_ + c * H_;     // c==16 -> col D_ (k) exactly
    const float x1 = p[j];
    const float x2 = p[j + 64];
    const float sn = sinT[t * H_ + j];
    const float cs = cosT[t * H_ + j];
    p[j]      = x1 * cs - x2 * sn;
    p[j + 64] = x2 * cs + x1 * sn;
}

// ---------------------------------------------------------------------------
// Flash-style causal MQA: one block per (b, head, 64-query tile).
// 128 threads = 4 waves; each wave owns 16 query rows.
// K staged row-major (async B128 when available) -> K^T fragments contiguous.
// V staged TRANSPOSED -> P@V B-fragments are single ds_load_b64.
// ---------------------------------------------------------------------------
__global__ __launch_bounds__(128) void mqa_attn_kernel(const float* __restrict__ qkv,
                                                       float* __restrict__ attn) {
    __shared__ float Ks[64][132];    // row-major keys (K^T frags contiguous in k)
    __shared__ float Vt[128][66];    // values transposed: [h][s]
    __shared__ float Ps[4][16][68];  // per-wave P re-layout pad

    const int tid  = threadIdx.x;
    const int lane = tid & 31;
    const int wid  = tid >> 5;
    const int lh   = lane >> 4;
    const int ll   = lane & 15;

    const int qt = blockIdx.x & 31;           // T/64 = 32 query tiles
    const int n  = (blockIdx.x >> 5) & 15;
    const int b  = blockIdx.x >> 9;
    const int q0 = qt * 64;

    const float scale = 0.08838834764831845f; // 1/sqrt(H)

    // Q A-fragments in registers, pre-scaled: lane = row % 16, K pair by lane half
    v2f aq[32];
    {
        const float* qp = qkv + (size_t)(b * T_ + q0 + wid * 16 + ll) * DQKV_ + n * H_;
        #pragma unroll
        for (int s = 0; s < 32; ++s) {
            const int k = s * 4 + 2 * lh;
            aq[s].x = qp[k] * scale;
            aq[s].y = qp[k + 1] * scale;
        }
    }

    v8f   o[8] = {};
    float mrow[8], lrow[8];
    #pragma unroll
    for (int v = 0; v < 8; ++v) { mrow[v] = -3.0e38f; lrow[v] = 0.0f; }

    for (int kt = 0; kt <= qt; ++kt) {
        // stage K rows async; V transposed via registers
        #pragma unroll
        for (int i = 0; i < 16; ++i) {
            const int fi = tid + 128 * i;
            const int r  = fi >> 5;
            const int c4 = (fi & 31) << 2;
            const size_t base = (size_t)(b * T_ + kt * 64 + r) * DQKV_;
#if USE_ASYNC_LDS
            ASYNC_B128(&qkv[base + D_ + c4], &Ks[r][c4]);
#else
            *(float4*)&Ks[r][c4] = *(const float4*)&qkv[base + D_ + c4];
#endif
            const float4 v = *(const float4*)&qkv[base + D_ + H_ + c4];
            Vt[c4 + 0][r] = v.x;
            Vt[c4 + 1][r] = v.y;
            Vt[c4 + 2][r] = v.z;
            Vt[c4 + 3][r] = v.w;
        }
#if USE_ASYNC_LDS
        __builtin_amdgcn_s_wait_asynccnt(0);
#endif
        __syncthreads();

        // S = Qscaled @ K^T : 16x64 per wave (4 col tiles)
        v8f sc[4] = {};
        #pragma unroll
        for (int s = 0; s < 32; ++s) {
            const int k = s * 4 + 2 * lh;
            #pragma unroll
            for (int c = 0; c < 4; ++c) {
                const v2f bk = *(const v2f*)&Ks[c * 16 + ll][k];  // B[k][s] = K[s][k]
                sc[c] = wmma_f32(aq[s], bk, sc[c]);
            }
        }

        if (kt == qt) {   // causal mask inside the diagonal tile (uniform branch)
            #pragma unroll
            for (int c = 0; c < 4; ++c)
                #pragma unroll
                for (int v = 0; v < 8; ++v) {
                    const int qr = wid * 16 + v + 8 * lh;  // row within 64-tile
                    const int cc = c * 16 + ll;            // col within 64-tile
                    if (cc > qr) sc[c][v] = -3.0e38f;
                }
        }

        // online softmax: row stats live per (VGPR index, lane half)
        float mnew[8], alpha[8];
        #pragma unroll
        for (int v = 0; v < 8; ++v) {
            float rm = fmaxf(fmaxf(sc[0][v], sc[1][v]), fmaxf(sc[2][v], sc[3][v]));
            #pragma unroll
            for (int m = 8; m >= 1; m >>= 1)
                rm = fmaxf(rm, __shfl_xor(rm, m, 32));   // reduces within 16-lane halves
            mnew[v]  = fmaxf(mrow[v], rm);
            alpha[v] = __expf(mrow[v] - mnew[v]);
            mrow[v]  = mnew[v];
        }
        #pragma unroll
        for (int c = 0; c < 4; ++c)
            #pragma unroll
            for (int v = 0; v < 8; ++v)
                sc[c][v] = __expf(sc[c][v] - mnew[v]);
        #pragma unroll
        for (int v = 0; v < 8; ++v) {
            float rs = sc[0][v] + sc[1][v] + sc[2][v] + sc[3][v];
            #pragma unroll
            for (int m = 8; m >= 1; m >>= 1)
                rs += __shfl_xor(rs, m, 32);
            lrow[v] = lrow[v] * alpha[v] + rs;
        }
        #pragma unroll
        for (int h = 0; h < 8; ++h)
            #pragma unroll
            for (int v = 0; v < 8; ++v)
                o[h][v] *= alpha[v];

        // P: C-layout -> row-major LDS (per-wave pad), then consume as A-fragments
        #pragma unroll
        for (int c = 0; c < 4; ++c)
            #pragma unroll
            for (int v = 0; v < 8; ++v)
                Ps[wid][v + 8 * lh][c * 16 + ll] = sc[c][v];
        __syncthreads();

        // O += P(16x64) @ V(64x128); V fragments contiguous in Vt
        #pragma unroll
        for (int s = 0; s < 16; ++s) {
            const int k = s * 4 + 2 * lh;
            const v2f ap = *(const v2f*)&Ps[wid][ll][k];
            #pragma unroll
            for (int h = 0; h < 8; ++h) {
                const v2f bv = *(const v2f*)&Vt[h * 16 + ll][k];  // B[k][h] = V[k][h]
                o[h] = wmma_f32(ap, bv, o[h]);
            }
        }
        __syncthreads();   // protect Ks/Vt before next tile's staging
    }

    // normalize and write attn out as (B,T,D) with head n at cols [n*H, n*H+H)
    float rinv[8];
    #pragma unroll
    for (int v = 0; v < 8; ++v) rinv[v] = 1.0f / lrow[v];
    #pragma unroll
    for (int h = 0; h < 8; ++h)
        #pragma unroll
        for (int v = 0; v < 8; ++v) {
            const size_t row = (size_t)(b * T_ + q0 + wid * 16 + v + 8 * lh);
            attn[row * D_ + n * H_ + h * 16 + ll] = o[h][v] * rinv[v];
        }
}

// ---------------------------------------------------------------------------
extern "C" void kernel_launch(void* const* d_in, const int* in_sizes, int n_in,
                              void* d_out, int out_size, void* d_ws, size_t ws_size,
                              hipStream_t stream) {
    (void)in_sizes; (void)n_in; (void)out_size; (void)ws_size;

    const float* x    = (const float*)d_in[0];  // (B,T,D)
    const float* sinT = (const float*)d_in[1];  // (T,H)
    const float* cosT = (const float*)d_in[2];  // (T,H)
    const float* Wqkv = (const float*)d_in[3];  // (D, D+2H)
    const float* Wout = (const float*)d_in[4];  // (D, D)
    float* out  = (float*)d_out;                // (B,T,D)

    float* qkv  = (float*)d_ws;                         // (B*T, DQKV)  ~37.7 MB
    float* attn = qkv + (size_t)B_ * T_ * DQKV_;        // (B*T, D)     ~33.6 MB

    // 1) qkv = x @ W_qkv
    gemm128_f32<<<dim3(DQKV_ / 128, (B_ * T_) / 128), 256, 0, stream>>>(
        x, Wqkv, qkv, B_ * T_, DQKV_, D_);

    // 2) RoPE on q heads and shared k (in place)
    rope_kernel<<<(B_ * T_ * 64 * (N_ + 1)) / 256, 256, 0, stream>>>(qkv, sinT, cosT);

    // 3) causal MQA attention -> attn (B,T,D)
    mqa_attn_kernel<<<B_ * N_ * (T_ / 64), 128, 0, stream>>>(qkv, attn);

    // 4) out = attn @ W_out
    gemm128_f32<<<dim3(D_ / 128, (B_ * T_) / 128), 256, 0, stream>>>(
        attn, Wout, out, B_ * T_, D_, D_);
}